// LSTMModule_8546984919149
// MI455X (gfx1250) — compile-verified
//
#include <hip/hip_runtime.h>

typedef __attribute__((ext_vector_type(2))) float v2f;
typedef __attribute__((ext_vector_type(8))) float v8f;

#define BB     32      // batch
#define SS     2048    // seq len
#define DD     512     // input dim
#define HH     512     // hidden dim
#define GG     2048    // 4*H gate width
#define KD     1024    // D+H == H+H (GEMM K)
#define NGROUP 2       // batch groups of 16
#define MBATCH 16      // batches per group (== WMMA M)
#define NSLAB  16      // hidden-column slabs per group
#define JS     32      // hidden units per slab
#define NBLK   16      // blocks per sync group
#define GSTR   132     // padded gates-LDS row stride

__device__ __forceinline__ float fast_rcp(float x) {
    return __builtin_amdgcn_rcpf(x);          // v_rcp_f32, ~1 ulp
}
__device__ __forceinline__ float sigf(float x) {
    return fast_rcp(1.0f + __expf(-x));
}
__device__ __forceinline__ float tanh_fast(float x) {
    x = fminf(15.0f, fmaxf(-15.0f, x));
    float e = __expf(2.0f * x);
    return (e - 1.0f) * fast_rcp(e + 1.0f);
}

// spin-wait until *p >= v (device scope); one lane spins, rest park at barrier
__device__ __forceinline__ void wait_ge(unsigned* p, unsigned v) {
    if (threadIdx.x == 0) {
        while (__hip_atomic_load(p, __ATOMIC_ACQUIRE, __HIP_MEMORY_SCOPE_AGENT) < v) {
            __builtin_amdgcn_s_sleep(1);
        }
    }
    __syncthreads();
}

__device__ __forceinline__ void signal_done(unsigned* p) {
    __threadfence();          // make this thread's global stores visible (agent)
    __syncthreads();          // all threads in block have fenced
    if (threadIdx.x == 0) {
        __hip_atomic_fetch_add(p, 1u, __ATOMIC_RELEASE, __HIP_MEMORY_SCOPE_AGENT);
    }
}

// One 16x16 fp32 C tile over half of K (512) with V_WMMA_F32_16X16X4_F32.
// Two independent accumulator chains (even/odd k-quads) to pipeline the XDL.
// Aptr: per-lane pointer to row m of the A operand (contiguous K).
// Wb:   W + khalf*512*GG + colg  (per-lane column pointer, row stride GG).
__device__ __forceinline__ v8f gemm_half(const float* __restrict__ Aptr,
                                         const float* __restrict__ Wb,
                                         int koff) {
    v8f acc0 = {};
    v8f acc1 = {};
#pragma unroll 4
    for (int k = 0; k < KD / 2; k += 8) {
        {
            const int kk = k + koff;                // lanes 0-15: k,k+1 ; 16-31: k+2,k+3
            float2 a2 = *(const float2*)(Aptr + kk);
            v2f av; av.x = a2.x; av.y = a2.y;
            v2f bv;
            bv.x = Wb[(size_t)kk * GG];
            bv.y = Wb[(size_t)kk * GG + GG];
            acc0 = __builtin_amdgcn_wmma_f32_16x16x4_f32(false, av, false, bv,
                                                         (short)0, acc0, false, false);
        }
        {
            const int kk = k + 4 + koff;
            float2 a2 = *(const float2*)(Aptr + kk);
            v2f av; av.x = a2.x; av.y = a2.y;
            v2f bv;
            bv.x = Wb[(size_t)kk * GG];
            bv.y = Wb[(size_t)kk * GG + GG];
            acc1 = __builtin_amdgcn_wmma_f32_16x16x4_f32(false, av, false, bv,
                                                         (short)0, acc1, false, false);
        }
    }
#pragma unroll
    for (int r = 0; r < 8; ++r) acc0[r] += acc1[r];
    return acc0;
}

extern "C" __global__ void init_ws_kernel(unsigned* p, int n) {
    int i = blockIdx.x * blockDim.x + threadIdx.x;
    if (i < n) p[i] = 0u;
}

extern "C" __global__ void __launch_bounds__(512, 1)
lstm_persist_kernel(const float* __restrict__ x, const float* __restrict__ W0,
                    const float* __restrict__ b0, const float* __restrict__ W1,
                    const float* __restrict__ b1, float* __restrict__ out,
                    float* __restrict__ h0buf, float* __restrict__ h1buf,
                    unsigned* done0, unsigned* done1) {
    // gates partial sums: [khalf][batch row][gate*32 + j]
    __shared__ float Glds[2][MBATCH][GSTR];

    const int tid   = threadIdx.x;
    const int lane  = tid & 31;
    const int wave  = tid >> 5;       // 16 waves
    const int nt    = wave & 7;       // 8 N-tiles of 16 cols
    const int khalf = wave >> 3;      // K split in half across wave pairs
    const int m     = lane & 15;      // A/C row within tile, B/C col
    const int koff  = (lane >> 4) << 1;

    const int slab  = blockIdx.x & (NSLAB - 1);
    const int group = blockIdx.x >> 4;   // 0..1

    // this wave's global W column: gate-block layout (i,f,g,o each HH wide)
    const int gate_w = nt >> 1;
    const int colg   = gate_w * HH + slab * JS + (nt & 1) * 16 + m;

    // cell phase: one (batch, local j) pair per thread
    const int cb = tid >> 5;             // 0..15
    const int cj = tid & 31;             // 0..31
    const int gb = group * MBATCH + cb;  // global batch
    const int jg = slab * JS + cj;       // global hidden index

    // hoist loop-invariant biases into registers (fences otherwise block hoisting)
    const float bi0 = b0[0 * HH + jg], bf0 = b0[1 * HH + jg];
    const float bg0 = b0[2 * HH + jg], bo0 = b0[3 * HH + jg];
    const float bi1 = b1[0 * HH + jg], bf1 = b1[1 * HH + jg];
    const float bg1 = b1[2 * HH + jg], bo1 = b1[3 * HH + jg];

    unsigned* d0 = done0 + group * SS;
    unsigned* d1 = done1 + group * SS;

    const float* xRowA = x + (((size_t)(group * MBATCH + m)) * SS) * DD; // + t*DD
    const size_t hRowA = (size_t)(group * MBATCH + m) * HH;

    float cc0 = 0.f, cc1 = 0.f, hh0 = 0.f, hh1 = 0.f;

    for (int t = 0; t < SS; ++t) {
        const size_t bufCur  = (size_t)(t & 1) * BB * HH;
        const size_t bufPrev = (size_t)((t & 1) ^ 1) * BB * HH;

        // ================= layer 0: gates = [x_t | h0_prev] @ W0 =================
        if (t > 0) wait_ge(&d0[t - 1], NBLK);   // h0[t-1] ready everywhere; safe to overwrite h0[t-2]

        if (t + 1 < SS) {   // prefetch next timestep's x slab (2KB per batch row)
            const char* nx = (const char*)(x + (((size_t)(group * MBATCH + (tid >> 5))) * SS
                                               + (size_t)(t + 1)) * DD)
                             + (size_t)(tid & 31) * 64;
            __builtin_prefetch(nx, 0, 1);
        }

        {
            const float* Aptr = (khalf == 0)
                ? (xRowA + (size_t)t * DD)                 // K 0..511  <- x_t
                : (h0buf + bufPrev + hRowA);               // K 512..1023 <- h0_prev
            const float* Wb = W0 + (size_t)(khalf * (KD / 2)) * GG + colg;
            v8f acc = gemm_half(Aptr, Wb, koff);
            const int rbase = (lane >> 4) << 3;            // C: M=r (lanes 0-15), M=r+8 (16-31)
#pragma unroll
            for (int r = 0; r < 8; ++r)
                Glds[khalf][rbase + r][nt * 16 + m] = acc[r];
        }
        __syncthreads();
        {
            float ip = Glds[0][cb][0 * JS + cj] + Glds[1][cb][0 * JS + cj] + bi0;
            float fp = Glds[0][cb][1 * JS + cj] + Glds[1][cb][1 * JS + cj] + bf0;
            float gp = Glds[0][cb][2 * JS + cj] + Glds[1][cb][2 * JS + cj] + bg0;
            float op = Glds[0][cb][3 * JS + cj] + Glds[1][cb][3 * JS + cj] + bo0;
            cc0 = sigf(fp) * cc0 + sigf(ip) * tanh_fast(gp);
            hh0 = sigf(op) * tanh_fast(cc0);
            h0buf[bufCur + (size_t)gb * HH + jg] = hh0;
        }
        signal_done(&d0[t]);

        // ================= layer 1: gates = [h0_t | h1_prev] @ W1 =================
        wait_ge(&d0[t], NBLK);                    // full h0[t] visible
        if (t > 0) wait_ge(&d1[t - 1], NBLK);     // full h1[t-1] visible; safe to overwrite h1[t-2]
        {
            const float* Aptr = (khalf == 0)
                ? (h0buf + bufCur + hRowA)                 // K 0..511  <- h0_t
                : (h1buf + bufPrev + hRowA);               // K 512..1023 <- h1_prev
            const float* Wb = W1 + (size_t)(khalf * (KD / 2)) * GG + colg;
            v8f acc = gemm_half(Aptr, Wb, koff);
            const int rbase = (lane >> 4) << 3;
#pragma unroll
            for (int r = 0; r < 8; ++r)
                Glds[khalf][rbase + r][nt * 16 + m] = acc[r];
        }
        __syncthreads();
        {
            float ip = Glds[0][cb][0 * JS + cj] + Glds[1][cb][0 * JS + cj] + bi1;
            float fp = Glds[0][cb][1 * JS + cj] + Glds[1][cb][1 * JS + cj] + bf1;
            float gp = Glds[0][cb][2 * JS + cj] + Glds[1][cb][2 * JS + cj] + bg1;
            float op = Glds[0][cb][3 * JS + cj] + Glds[1][cb][3 * JS + cj] + bo1;
            cc1 = sigf(fp) * cc1 + sigf(ip) * tanh_fast(gp);
            hh1 = sigf(op) * tanh_fast(cc1);
            h1buf[bufCur + (size_t)gb * HH + jg] = hh1;
            out[((size_t)gb * SS + (size_t)t) * HH + jg] = hh1;   // outs[b][t][j]
        }
        signal_done(&d1[t]);
    }

    // final states: outs | h0 | c0 | h1 | c1
    {
        const size_t base = (size_t)BB * SS * HH;
        const size_t idx  = (size_t)gb * HH + jg;
        out[base + 0 * (size_t)BB * HH + idx] = hh0;
        out[base + 1 * (size_t)BB * HH + idx] = cc0;
        out[base + 2 * (size_t)BB * HH + idx] = hh1;
        out[base + 3 * (size_t)BB * HH + idx] = cc1;
    }
}

extern "C" void kernel_launch(void* const* d_in, const int* in_sizes, int n_in,
                              void* d_out, int out_size, void* d_ws, size_t ws_size,
                              hipStream_t stream) {
    (void)in_sizes; (void)n_in; (void)out_size; (void)ws_size;
    const float* x  = (const float*)d_in[0];
    const float* W0 = (const float*)d_in[1];
    const float* b0 = (const float*)d_in[2];
    const float* W1 = (const float*)d_in[3];
    const float* b1 = (const float*)d_in[4];
    float* out = (float*)d_out;

    // workspace layout (dwords): done0[2*2048] | done1[2*2048] | h0buf[2*32*512] | h1buf[2*32*512]
    unsigned* done0 = (unsigned*)d_ws;
    unsigned* done1 = done0 + NGROUP * SS;
    float* h0buf = (float*)(done1 + NGROUP * SS);
    float* h1buf = h0buf + 2 * BB * HH;

    const int initN = 2 * NGROUP * SS + 2 * (2 * BB * HH);  // 73728 dwords
    init_ws_kernel<<<(initN + 255) / 256, 256, 0, stream>>>((unsigned*)d_ws, initN);

    lstm_persist_kernel<<<NGROUP * NSLAB, 512, 0, stream>>>(
        x, W0, b0, W1, b1, out, h0buf, h1buf, done0, done1);
}